// CapsuleConv2d_83064667504742
// MI455X (gfx1250) — compile-verified
//
#include <hip/hip_runtime.h>

typedef __attribute__((ext_vector_type(2))) float v2f;
typedef __attribute__((ext_vector_type(8))) float v8f;

#define O_CAPS 8
#define L_DIM  16
#define P_DIM  8
#define F_DIM  72
#define FPW    9     // f-values per wave; f = wave*9 + t  =>  g = wave, k = t
#define NWAVES 8
#define H_DIM  32
#define W_DIM  32
#define HW     1024  // 32*32

__launch_bounds__(256, 1)
__global__ void capsule_conv_route(const float* __restrict__ x,
                                   const float* __restrict__ weight,
                                   float* __restrict__ out) {
    __shared__ float s_part[NWAVES][256]; // per-wave partial s[l][w]
    __shared__ float s_red[256];          // reduced s[l][w], scaled by 1/G

    const int tid  = threadIdx.x;
    const int lane = tid & 31;
    const int wave = tid >> 5;
    const int lo16 = lane & 15;
    const int hi   = lane >> 4;   // 0: lanes 0-15, 1: lanes 16-31
    const int koff = hi * 2;      // K offset inside a 4-wide chunk (ISA A/B f32 layout)

    // block -> (n, o, spatial tile of 16 contiguous x's)
    const int bid = blockIdx.x;
    const int n   = bid >> 9;          // / (8*64)
    const int o   = (bid >> 6) & 7;
    const int sp  = bid & 63;
    const int y   = sp >> 1;
    const int x0  = (sp & 1) << 4;
    const int xxb = x0 + lo16;         // spatial column this lane owns (N index of B/C)

    // g = wave: this wave's input-capsule group; channel base plane
    const float* xg = x + (size_t)n * 64 * HW + (size_t)(wave * P_DIM) * HW;
    // weight base for this wave's 9 f's: weight[o, wave*9 + t, l=lo16, p]
    const float* wb0 = weight
        + (((size_t)(o * F_DIM + wave * FPW) * L_DIM + lo16) * P_DIM) + koff;

    // ---- priors: 9 WMMA C-tiles per wave, kept in registers (72 VGPRs) ----
    v8f pr[FPW];
#pragma unroll
    for (int t = 0; t < FPW; ++t) {
        // k = t: kernel offset is a compile-time constant after unrolling
        const int dy = t / 3 - 1;
        const int dx = t % 3 - 1;
        const int yy = y + dy;
        const int xx = xxb + dx;
        const bool inb = (yy >= 0) & (yy < H_DIM) & (xx >= 0) & (xx < W_DIM);
        const float m = inb ? 1.0f : 0.0f;
        // clamp into range so the load is always legal; mask kills OOB values
        const int yyc = min(max(yy, 0), H_DIM - 1);
        const int xxc = min(max(xx, 0), W_DIM - 1);

        const float* wbase = wb0 + (size_t)t * L_DIM * P_DIM;
        const float* xrow  = xg + yyc * W_DIM + xxc;

        v8f c = {};
#pragma unroll
        for (int kb = 0; kb < P_DIM; kb += 4) {
            v2f a = *(const v2f*)(wbase + kb);                 // aligned float2
            v2f b;
            b.x = xrow[(size_t)(kb + koff) * HW] * m;          // channel kb+koff
            b.y = xrow[(size_t)(kb + koff + 1) * HW] * m;      // channel kb+koff+1
            // D(16x16,f32) = A(16x4,f32) x B(4x16,f32) + C
            c = __builtin_amdgcn_wmma_f32_16x16x4_f32(
                    false, a, false, b, (short)0, c, false, false);
        }
        pr[t] = c;
    }

    // ---- dynamic routing (3 iterations), priors never leave registers ----
    float probs[FPW];
#pragma unroll
    for (int t = 0; t < FPW; ++t) probs[t] = 1.0f;

    v8f outv = {};
#pragma unroll 1
    for (int r = 0; r < 3; ++r) {
        // per-wave partial s[l,w] = sum over this wave's f of probs*priors
        v8f ps = {};
#pragma unroll
        for (int t = 0; t < FPW; ++t) ps += probs[t] * pr[t];

        __syncthreads();                 // protect s_part reuse across iterations
#pragma unroll
        for (int j = 0; j < 8; ++j)
            s_part[wave][(hi * 8 + j) * 16 + lo16] = ps[j];
        __syncthreads();

        // deterministic cross-wave reduction: one (l,w) cell per thread
        {
            float acc = 0.0f;
#pragma unroll
            for (int wv = 0; wv < NWAVES; ++wv) acc += s_part[wv][tid];
            s_red[tid] = acc * 0.125f;   // mean over G=8
        }
        __syncthreads();

        // squash over l for this lane's column w = lo16
        float sq = 0.0f;
        float sv[16];
#pragma unroll
        for (int l = 0; l < 16; ++l) {
            float v = s_red[l * 16 + lo16];
            sv[l] = v;
            sq += v * v;
        }
        const float fac = sq / (1.0f + sq) * rsqrtf(sq);
#pragma unroll
        for (int j = 0; j < 8; ++j) outv[j] = sv[hi * 8 + j] * fac;

        if (r < 2) {
            // delta[f,w] = sum_l priors[f,l,w]*out[l,w]; each half-lane has 8 l's
#pragma unroll
            for (int t = 0; t < FPW; ++t) {
                float dh = 0.0f;
#pragma unroll
                for (int j = 0; j < 8; ++j) dh += pr[t][j] * outv[j];
                float d = dh + __shfl_xor(dh, 16, 32);   // wave32 partner half
                probs[t] += expf(d);
            }
        }
    }

    // ---- store: out[n, o*16 + l, y, x0 + lo16] (wave 0 only; outv identical) ----
    if (wave == 0) {
        float* op = out + ((size_t)(n * O_CAPS + o) * L_DIM) * HW
                        + y * W_DIM + x0 + lo16;
#pragma unroll
        for (int j = 0; j < 8; ++j) {
            const int l = hi * 8 + j;
            op[(size_t)l * HW] = outv[j];
        }
    }
}

extern "C" void kernel_launch(void* const* d_in, const int* in_sizes, int n_in,
                              void* d_out, int out_size, void* d_ws, size_t ws_size,
                              hipStream_t stream) {
    const float* x = (const float*)d_in[0];
    const float* w = (const float*)d_in[1];
    float* out = (float*)d_out;
    // 4 n * 8 o * 64 spatial tiles = 2048 blocks, 8 waves each
    capsule_conv_route<<<dim3(2048), dim3(256), 0, stream>>>(x, w, out);
}